// SE3TransformerLayer_37220186587477
// MI455X (gfx1250) — compile-verified
//
#include <hip/hip_runtime.h>
#include <hip/hip_bf16.h>

#define N_NODES 50000
#define DIM     64
#define N_EDGES 800000

typedef __attribute__((ext_vector_type(2))) float v2f;
typedef __attribute__((ext_vector_type(8))) float v8f;

// ---------------------------------------------------------------------------
// Kernel 1: node-level QKV projection, fp32 WMMA (V_WMMA_F32_16X16X4_F32).
// Q = x @ Wq^T + bq, etc.  One 128-thread block (4 wave32) per 16-row slab.
// Wave w computes output columns [16w, 16w+16) for Q, K and V.
// A layout (16x4 f32):  lane<16: (K=k0,k0+1), lane>=16: (K=k0+2,k0+3), M=lane&15
// B layout (4x16 f32):  mirrored (rows K across the 2 VGPR halves, N=lane&15)
// C/D layout (16x16):   vgpr r -> M = r + 8*(lane>=16), N = lane&15
// ---------------------------------------------------------------------------
__global__ __launch_bounds__(128)
void qkv_kernel(const float* __restrict__ x,
                const float* __restrict__ Wq, const float* __restrict__ bq,
                const float* __restrict__ Wk, const float* __restrict__ bk,
                const float* __restrict__ Wv, const float* __restrict__ bv,
                float* __restrict__ Q, float* __restrict__ K, float* __restrict__ V)
{
    __shared__ float xs[16][68];               // pad 64->68 (stride 4 mod 64 banks)
    const int slab = blockIdx.x;
    const int tid  = threadIdx.x;

    // cooperative LDS stage of the 16x64 x-slab (float4 moves)
    {
        const float4* src = (const float4*)(x + (size_t)slab * 16 * DIM);
        for (int c = tid; c < 16 * 16; c += 128) {
            const int r = c >> 4, q4 = c & 15;
            float4 val = src[c];
            *(float4*)&xs[r][q4 * 4] = val;
        }
    }
    __syncthreads();

    const int lane  = tid & 31;
    const int wave  = tid >> 5;
    const int h     = lane >> 4;               // half-wave select
    const int m     = lane & 15;               // A-row / B-col / D-col index
    const int jbase = wave * 16;               // output column tile

    v8f cq = {}; v8f ck = {}; v8f cv = {};
    #pragma unroll
    for (int kt = 0; kt < 16; ++kt) {
        const int k0 = kt * 4 + 2 * h;
        v2f a;  a.x  = xs[m][k0];              a.y  = xs[m][k0 + 1];
        const int jr = (jbase + m) * DIM + k0; // W row-major: W[j][k] = W^T[k][j]
        v2f bm; bm.x = Wq[jr];                 bm.y = Wq[jr + 1];
        cq = __builtin_amdgcn_wmma_f32_16x16x4_f32(false, a, false, bm,
                                                   (short)0, cq, false, false);
        bm.x = Wk[jr];                         bm.y = Wk[jr + 1];
        ck = __builtin_amdgcn_wmma_f32_16x16x4_f32(false, a, false, bm,
                                                   (short)0, ck, false, false);
        bm.x = Wv[jr];                         bm.y = Wv[jr + 1];
        cv = __builtin_amdgcn_wmma_f32_16x16x4_f32(false, a, false, bm,
                                                   (short)0, cv, false, false);
    }

    const int col = jbase + m;
    const float biq = bq[col], bik = bk[col], biv = bv[col];
    #pragma unroll
    for (int r = 0; r < 8; ++r) {
        const int row = slab * 16 + r + 8 * h;
        Q[(size_t)row * DIM + col] = cq[r] + biq;
        K[(size_t)row * DIM + col] = ck[r] + bik;
        V[(size_t)row * DIM + col] = cv[r] + biv;
    }
}

// ---------------------------------------------------------------------------
// ws ctrl init: max-key = ordered-uint(-inf), sum = 0
// ---------------------------------------------------------------------------
__global__ void init_ctrl(unsigned* __restrict__ maxk, float* __restrict__ sum)
{
    *maxk = 0x007FFFFFu;    // ~bits(-inf)
    *sum  = 0.0f;
}

// ---------------------------------------------------------------------------
// Kernel 2: per-edge logit attn[e] = Q[row]·K[col]; global max via atomicMax
// on an order-preserving uint encoding of float.
// ---------------------------------------------------------------------------
__global__ __launch_bounds__(256)
void attn_kernel(const float* __restrict__ Q, const float* __restrict__ K,
                 const long long* __restrict__ ei,
                 float* __restrict__ attn, unsigned* __restrict__ maxk)
{
    const int e   = blockIdx.x * 256 + threadIdx.x;
    const int row = (int)ei[e];
    const int col = (int)ei[N_EDGES + e];
    const float4* qp = (const float4*)(Q + (size_t)row * DIM);
    const float4* kp = (const float4*)(K + (size_t)col * DIM);
    float acc = 0.0f;
    #pragma unroll
    for (int i = 0; i < 16; ++i) {
        const float4 a = qp[i], b = kp[i];
        acc += a.x * b.x + a.y * b.y + a.z * b.z + a.w * b.w;
    }
    attn[e] = acc;

    float mv = acc;
    #pragma unroll
    for (int off = 16; off; off >>= 1)
        mv = fmaxf(mv, __shfl_xor(mv, off, 32));
    if ((threadIdx.x & 31) == 0) {
        const unsigned bits = __float_as_uint(mv);
        const unsigned key  = (bits & 0x80000000u) ? ~bits : (bits | 0x80000000u);
        atomicMax(maxk, key);
    }
}

// ---------------------------------------------------------------------------
// Kernel 3: attn[e] <- exp(attn[e] - max); accumulate softmax denominator.
// ---------------------------------------------------------------------------
__global__ __launch_bounds__(256)
void expsum_kernel(float* __restrict__ attn, const unsigned* __restrict__ maxk,
                   float* __restrict__ sum)
{
    const unsigned key = *maxk;
    const float mv = (key & 0x80000000u) ? __uint_as_float(key & 0x7FFFFFFFu)
                                         : __uint_as_float(~key);
    const int e = blockIdx.x * 256 + threadIdx.x;
    const float w = expf(attn[e] - mv);
    attn[e] = w;

    float s = w;
    #pragma unroll
    for (int off = 16; off; off >>= 1)
        s += __shfl_xor(s, off, 32);
    if ((threadIdx.x & 31) == 0) atomicAdd(sum, s);
}

// ---------------------------------------------------------------------------
// Kernel 4: out = concat(x, coord)  (d_out is poisoned; must be seeded)
// ---------------------------------------------------------------------------
__global__ __launch_bounds__(256)
void out_init_kernel(const float* __restrict__ x, const float* __restrict__ coord,
                     float* __restrict__ out)
{
    const int i  = blockIdx.x * 256 + threadIdx.x;        // float4 index
    const int Q4 = N_NODES * DIM / 4;
    float4* o = (float4*)out;
    o[i] = (i < Q4) ? ((const float4*)x)[i] : ((const float4*)coord)[i - Q4];
}

// ---------------------------------------------------------------------------
// Kernel 5: one wave32 per edge; lane owns dims d and d+32.
// delta_x    = w * V[col]
// delta_coor = w * (dist*Wc + bc) * (coord[row]-coord[col])
// scatter-added into out with f32 atomics.
// ---------------------------------------------------------------------------
__global__ __launch_bounds__(256)
void scatter_kernel(const float* __restrict__ coord, const float* __restrict__ V,
                    const float* __restrict__ Wc, const float* __restrict__ bc,
                    const long long* __restrict__ ei, const float* __restrict__ attn,
                    const float* __restrict__ sum, float* __restrict__ out)
{
    const float inv  = 1.0f / (*sum);
    const int lane   = threadIdx.x & 31;
    const int e      = blockIdx.x * 8 + (threadIdx.x >> 5);
    const int row    = (int)ei[e];
    const int col    = (int)ei[N_EDGES + e];
    const float w    = attn[e] * inv;
    const int d0 = lane, d1 = lane + 32;

    const float cr0 = coord[(size_t)row * DIM + d0];
    const float cr1 = coord[(size_t)row * DIM + d1];
    const float cc0 = coord[(size_t)col * DIM + d0];
    const float cc1 = coord[(size_t)col * DIM + d1];
    const float dx0 = cr0 - cc0, dx1 = cr1 - cc1;

    float loc = dx0 * dx0 + dx1 * dx1;
    #pragma unroll
    for (int off = 16; off; off >>= 1)
        loc += __shfl_xor(loc, off, 32);
    const float dist = sqrtf(loc);

    const float v0 = V[(size_t)col * DIM + d0];
    const float v1 = V[(size_t)col * DIM + d1];
    const float g0 = dist * Wc[d0] + bc[d0];
    const float g1 = dist * Wc[d1] + bc[d1];

    float* ox = out;
    float* oc = out + (size_t)N_NODES * DIM;
    atomicAdd(&ox[(size_t)row * DIM + d0], w * v0);
    atomicAdd(&ox[(size_t)row * DIM + d1], w * v1);
    atomicAdd(&oc[(size_t)row * DIM + d0], w * g0 * dx0);
    atomicAdd(&oc[(size_t)row * DIM + d1], w * g1 * dx1);
}

// ---------------------------------------------------------------------------
extern "C" void kernel_launch(void* const* d_in, const int* in_sizes, int n_in,
                              void* d_out, int out_size, void* d_ws, size_t ws_size,
                              hipStream_t stream)
{
    (void)in_sizes; (void)n_in; (void)out_size; (void)ws_size;

    const float*     x     = (const float*)d_in[0];
    const float*     coord = (const float*)d_in[1];
    const float*     Wq    = (const float*)d_in[2];
    const float*     bq    = (const float*)d_in[3];
    const float*     Wk    = (const float*)d_in[4];
    const float*     bk    = (const float*)d_in[5];
    const float*     Wv    = (const float*)d_in[6];
    const float*     bv    = (const float*)d_in[7];
    const float*     Wc    = (const float*)d_in[8];
    const float*     bc    = (const float*)d_in[9];
    const long long* ei    = (const long long*)d_in[10];
    float*           out   = (float*)d_out;

    // workspace: [ctrl 256B][attn E f32][Q N*64][K N*64][V N*64]  ~41.6 MB
    char*     ws   = (char*)d_ws;
    unsigned* maxk = (unsigned*)ws;
    float*    sum  = (float*)(ws + 4);
    float*    attn = (float*)(ws + 256);
    float*    Q    = (float*)(ws + 256 + (size_t)N_EDGES * 4);
    float*    K    = Q + (size_t)N_NODES * DIM;
    float*    V    = K + (size_t)N_NODES * DIM;

    init_ctrl<<<1, 1, 0, stream>>>(maxk, sum);
    qkv_kernel<<<N_NODES / 16, 128, 0, stream>>>(x, Wq, bq, Wk, bk, Wv, bv, Q, K, V);
    attn_kernel<<<N_EDGES / 256, 256, 0, stream>>>(Q, K, ei, attn, maxk);
    expsum_kernel<<<N_EDGES / 256, 256, 0, stream>>>(attn, maxk, sum);
    out_init_kernel<<<(2 * N_NODES * DIM / 4) / 256, 256, 0, stream>>>(x, coord, out);
    scatter_kernel<<<N_EDGES / 8, 256, 0, stream>>>(coord, V, Wc, bc, ei, attn, sum, out);
}